// UWNeRFRGBRenderer_59648505806899
// MI455X (gfx1250) — compile-verified
//
#include <hip/hip_runtime.h>
#include <cstdint>
#include <cstddef>

#define S_TOT 128
#define WAVES_PER_BLOCK 4
#define BLOCK_THREADS (WAVES_PER_BLOCK * 32)
#define SLOTS 14   // float4 slots per lane per ray: deltas,dens + 3 each of rgb/veil/direct/back

#if defined(__has_builtin)
# if __has_builtin(__builtin_amdgcn_global_load_async_to_lds_b128)
#  define HAVE_ASYNC_LD 1
# endif
# if __has_builtin(__builtin_amdgcn_s_wait_asynccnt)
#  define HAVE_ASYNC_WAIT 1
# endif
# if __has_builtin(__builtin_amdgcn_update_dpp) && __has_builtin(__builtin_amdgcn_readlane)
#  define HAVE_DPP 1
# endif
#endif

// Builtin params are AS(1)/AS(3) pointers to a 16-byte int vector (probe-confirmed).
typedef int v4i __attribute__((vector_size(16)));
typedef __attribute__((address_space(1))) v4i* gas_v4i_p;
typedef __attribute__((address_space(3))) v4i* las_v4i_p;

// Async copy of 16 bytes per lane from global into LDS (GLOBAL_LOAD_ASYNC_TO_LDS_B128,
// tracked by ASYNCcnt).
__device__ __forceinline__ void async_copy16(const void* g, void* l) {
#if defined(HAVE_ASYNC_LD)
  __builtin_amdgcn_global_load_async_to_lds_b128((gas_v4i_p)g, (las_v4i_p)l, 0, 0);
#else
  unsigned long long ga = (unsigned long long)(uintptr_t)g;
  unsigned int la = (unsigned int)(uintptr_t)l;  // low 32 bits of flat LDS addr == LDS offset
  asm volatile("global_load_async_to_lds_b128 %0, %1, off"
               :: "v"(la), "v"(ga) : "memory");
#endif
}

#if defined(HAVE_ASYNC_WAIT)
# define WAIT_ASYNC(n) __builtin_amdgcn_s_wait_asynccnt(n)
#else
# define WAIT_ASYNC(n) asm volatile("s_wait_asynccnt %0" :: "n"(n) : "memory")
#endif

__device__ __forceinline__ float nan0(float x) { return (x == x) ? x : 0.0f; }
__device__ __forceinline__ float clamp01(float x) { return fminf(fmaxf(x, 0.0f), 1.0f); }

#if defined(HAVE_DPP)
// x += dpp_select(x) with zero-identity for shifted-in lanes (bound_ctrl).
template <int CTRL>
__device__ __forceinline__ float dpp_add(float x) {
  int s = __builtin_amdgcn_update_dpp(0, __float_as_int(x), CTRL, 0xf, 0xf, true);
  return x + __int_as_float(s);
}

__device__ __forceinline__ float lane_bcast(float x, int srclane) {
  return __int_as_float(__builtin_amdgcn_readlane(__float_as_int(x), srclane));
}

// Wave32 exclusive prefix sum via DPP row_shr chain + cross-row readlane fix.
__device__ __forceinline__ float wave_excl_scan(float tot, int lane) {
  float x = tot;
  x = dpp_add<0x111>(x);  // row_shr:1
  x = dpp_add<0x112>(x);  // row_shr:2
  x = dpp_add<0x114>(x);  // row_shr:4
  x = dpp_add<0x118>(x);  // row_shr:8
  const float r15 = lane_bcast(x, 15);  // row0 inclusive total
  if (lane >= 16) x += r15;
  return x - tot;
}

// Wave32 all-lanes sum via DPP xor-distance adds + two readlanes (uniform result).
__device__ __forceinline__ float wave_sum(float x) {
  x = dpp_add<0xB1>(x);   // quad_perm {1,0,3,2}  (xor 1)
  x = dpp_add<0x4E>(x);   // quad_perm {2,3,0,1}  (xor 2)
  x = dpp_add<0x141>(x);  // row_half_mirror      (xor 4)
  x = dpp_add<0x140>(x);  // row_mirror           (xor 8)
  return lane_bcast(x, 0) + lane_bcast(x, 16);
}
#else
__device__ __forceinline__ float lane_bcast(float x, int srclane) {
  return __shfl(x, srclane, 32);
}
__device__ __forceinline__ float wave_excl_scan(float tot, int lane) {
  float v = tot;
#pragma unroll
  for (int d = 1; d < 32; d <<= 1) {
    float o = __shfl_up(v, (unsigned)d, 32);
    if (lane >= d) v += o;
  }
  return v - tot;
}
__device__ __forceinline__ float wave_sum(float x) {
#pragma unroll
  for (int m = 16; m >= 1; m >>= 1) x += __shfl_xor(x, m, 32);
  return x;
}
#endif

__global__ __launch_bounds__(BLOCK_THREADS)
void uwnerf_render_kernel(const float* __restrict__ densities,
                          const float* __restrict__ rgbs,
                          const float* __restrict__ veiling,
                          const float* __restrict__ direct_c,
                          const float* __restrict__ back_c,
                          const float* __restrict__ deltas,
                          float* __restrict__ out, int nrays) {
  // Double-buffered per-wave staging: [buf][wave][slot][lane] -> bank-conflict-free
  // ds_load_b128 (consecutive lanes hit consecutive banks). 57,344 bytes per WG.
  __shared__ float4 stage[2][WAVES_PER_BLOCK][SLOTS][32];

  const int lane    = (int)(threadIdx.x & 31u);
  const int wave    = (int)(threadIdx.x >> 5u);
  const int gwave   = (int)blockIdx.x * WAVES_PER_BLOCK + wave;
  const int wstride = (int)gridDim.x * WAVES_PER_BLOCK;

  auto issue = [&](int ray, int b) {
    const size_t r1 = (size_t)ray * S_TOT;       // 1-channel arrays
    const size_t r3 = (size_t)ray * S_TOT * 3;   // 3-channel arrays
    async_copy16(deltas    + r1 + (size_t)lane * 4, &stage[b][wave][0][lane]);
    async_copy16(densities + r1 + (size_t)lane * 4, &stage[b][wave][1][lane]);
#pragma unroll
    for (int k = 0; k < 3; ++k) {
      const size_t off = r3 + (size_t)lane * 12 + (size_t)k * 4;
      async_copy16(rgbs     + off, &stage[b][wave][2 + k][lane]);
      async_copy16(veiling  + off, &stage[b][wave][5 + k][lane]);
      async_copy16(direct_c + off, &stage[b][wave][8 + k][lane]);
      async_copy16(back_c   + off, &stage[b][wave][11 + k][lane]);
    }
  };

  int buf = 0;
  if (gwave < nrays) issue(gwave, 0);

  for (int ray = gwave; ray < nrays; ray += wstride) {
    const int nxt = ray + wstride;
    if (nxt < nrays) {
      issue(nxt, buf ^ 1);
      WAIT_ASYNC(SLOTS);  // async loads complete in order: oldest 14 (current buf) done
    } else {
      WAIT_ASYNC(0);
    }

    // ---- read staged data (lane owns samples 4*lane .. 4*lane+3) ----
    const float4 fdel = stage[buf][wave][0][lane];
    const float4 fden = stage[buf][wave][1][lane];
    float4 q[12];
#pragma unroll
    for (int k = 0; k < 12; ++k) q[k] = stage[buf][wave][2 + k][lane];

    float del[4] = {fdel.x, fdel.y, fdel.z, fdel.w};
    float den[4] = {fden.x, fden.y, fden.z, fden.w};
    float rgbv[12], veilv[12], dcv[12], bcv[12];
#pragma unroll
    for (int k = 0; k < 3; ++k) {
      rgbv[4*k+0]  = nan0(q[k].x);   rgbv[4*k+1]  = nan0(q[k].y);
      rgbv[4*k+2]  = nan0(q[k].z);   rgbv[4*k+3]  = nan0(q[k].w);
      veilv[4*k+0] = nan0(q[3+k].x); veilv[4*k+1] = nan0(q[3+k].y);
      veilv[4*k+2] = nan0(q[3+k].z); veilv[4*k+3] = nan0(q[3+k].w);
      dcv[4*k+0]   = nan0(q[6+k].x); dcv[4*k+1]   = nan0(q[6+k].y);
      dcv[4*k+2]   = nan0(q[6+k].z); dcv[4*k+3]   = nan0(q[6+k].w);
      bcv[4*k+0]   = nan0(q[9+k].x); bcv[4*k+1]   = nan0(q[9+k].y);
      bcv[4*k+2]   = nan0(q[9+k].z); bcv[4*k+3]   = nan0(q[9+k].w);
    }

    // ---- density scan -> transmittance -> weights ----
    float odd[4]; float tot = 0.f;
#pragma unroll
    for (int i = 0; i < 4; ++i) { odd[i] = del[i] * den[i]; tot += odd[i]; }
    float pre = wave_excl_scan(tot, lane);
    float w[4];
#pragma unroll
    for (int i = 0; i < 4; ++i) {
      const float T = __expf(-pre);
      const float a = 1.0f - __expf(-odd[i]);
      w[i] = a * T;
      pre += odd[i];
    }

    float lws = 0.f;
#pragma unroll
    for (int i = 0; i < 4; ++i) lws += w[i];

    // ---- per-channel direct/backscatter scans + weighted accumulation ----
    float rs[3], dsum[3], bsum[3];
#pragma unroll
    for (int c = 0; c < 3; ++c) {
      float dd[4], bd[4]; float dt = 0.f, bt = 0.f;
#pragma unroll
      for (int i = 0; i < 4; ++i) {
        dd[i] = del[i] * dcv[i*3+c]; dt += dd[i];
        bd[i] = del[i] * bcv[i*3+c]; bt += bd[i];
      }
      float pd = wave_excl_scan(dt, lane);
      float pb = wave_excl_scan(bt, lane);
      float a_rs = 0.f, a_ds = 0.f, a_bs = 0.f;
#pragma unroll
      for (int i = 0; i < 4; ++i) {
        const float datt = __expf(-pd);
        const float batt = 1.0f - __expf(-pb);
        const float wr = w[i] * rgbv[i*3+c];
        a_rs += wr;
        a_ds += wr * datt;
        a_bs += w[i] * batt * veilv[i*3+c];
        pd += dd[i]; pb += bd[i];
      }
      rs[c] = a_rs; dsum[c] = a_ds; bsum[c] = a_bs;
    }

    // ---- wave-wide sums (DPP; uniform result on all lanes) ----
    const float ws = wave_sum(lws);
#pragma unroll
    for (int c = 0; c < 3; ++c) {
      rs[c]   = wave_sum(rs[c]);
      dsum[c] = wave_sum(dsum[c]);
      bsum[c] = wave_sum(bsum[c]);
    }

    // veiling_light[ray, 0, :] lives in lane 0's veilv[0..2]
    float vl0[3];
#pragma unroll
    for (int c = 0; c < 3; ++c) vl0[c] = lane_bcast(veilv[c], 0);

    if (lane == 0) {
      const float mask = clamp01(ws);
      const size_t o  = (size_t)ray * 3;
      const size_t R3 = (size_t)nrays * 3;
#pragma unroll
      for (int c = 0; c < 3; ++c) {
        const float med = (1.0f - mask) * vl0[c];
        const float bsr = bsum[c] + med;
        out[o + c]          = clamp01(dsum[c] + bsr);  // rgb
        out[R3 + o + c]     = clamp01(rs[c]);          // restored
        out[2*R3 + o + c]   = clamp01(dsum[c]);        // direct
        out[3*R3 + o + c]   = clamp01(bsr);            // backscatter
      }
    }
    buf ^= 1;
  }
}

extern "C" void kernel_launch(void* const* d_in, const int* in_sizes, int n_in,
                              void* d_out, int out_size, void* d_ws, size_t ws_size,
                              hipStream_t stream) {
  const float* densities = (const float*)d_in[0];
  const float* rgbs      = (const float*)d_in[1];
  const float* veiling   = (const float*)d_in[2];
  const float* direct_c  = (const float*)d_in[3];
  const float* back_c    = (const float*)d_in[4];
  const float* deltas    = (const float*)d_in[5];
  float* out = (float*)d_out;

  const int nrays = in_sizes[0] / S_TOT;  // densities is [R, S, 1]
  // ~4 rays per wave so the async double-buffering has loop iterations to hide.
  int total_waves = (nrays + 3) / 4;
  int blocks = (total_waves + WAVES_PER_BLOCK - 1) / WAVES_PER_BLOCK;
  if (blocks < 1) blocks = 1;

  hipLaunchKernelGGL(uwnerf_render_kernel, dim3(blocks), dim3(BLOCK_THREADS), 0, stream,
                     densities, rgbs, veiling, direct_c, back_c, deltas, out, nrays);
}